// Attention_15788299780825
// MI455X (gfx1250) — compile-verified
//
#include <hip/hip_runtime.h>
#include <hip/hip_bf16.h>

typedef __attribute__((ext_vector_type(2))) float v2f;
typedef __attribute__((ext_vector_type(8))) float v8f;

#define BB 16
#define CC 512
#define NN 4096
#define CR 128
#define BN_EPS 1e-5f

// ---------------------------------------------------------------------------
// Kernel 1: s[b,n] = sum_c x[b,c,n].  One thread per (b,n); each wave loads a
// contiguous 128B line per c-iteration (fully coalesced).  This is the one
// pass that must come from HBM (134 MB read).
// ---------------------------------------------------------------------------
__global__ __launch_bounds__(256) void colsum_kernel(
    const float* __restrict__ x, float* __restrict__ s) {
  const int n = blockIdx.x * 256 + threadIdx.x;
  const int b = blockIdx.y;
  const float* xp = x + (size_t)b * CC * NN + n;
  float acc = 0.0f;
#pragma unroll 8
  for (int c = 0; c < CC; ++c) acc += xp[(size_t)c * NN];
  s[b * NN + n] = acc;
}

// ---------------------------------------------------------------------------
// Kernel 2: att[b,c] = sum_n x[b,c,n] * s[b,n].  x should be L2-resident
// (134 MB < 192 MB L2); s is 256 KB.  One block per (b,c), LDS tree reduce.
// ---------------------------------------------------------------------------
__global__ __launch_bounds__(256) void att_kernel(
    const float* __restrict__ x, const float* __restrict__ s,
    float* __restrict__ att) {
  const int c = blockIdx.x;
  const int b = blockIdx.y;
  const float* xp = x + ((size_t)b * CC + c) * NN;
  const float* sp = s + b * NN;
  float acc = 0.0f;
#pragma unroll 4
  for (int i = threadIdx.x; i < NN; i += 256) acc += xp[i] * sp[i];
  __shared__ float red[256];
  red[threadIdx.x] = acc;
  __syncthreads();
  for (int w = 128; w > 0; w >>= 1) {
    if ((int)threadIdx.x < w) red[threadIdx.x] += red[threadIdx.x + w];
    __syncthreads();
  }
  if (threadIdx.x == 0) att[b * CC + c] = red[0];
}

// ---------------------------------------------------------------------------
// Kernel 3: fused  h = relu(BN(att @ W1^T)) ;  a = sigmoid(h @ W2^T)
// Single 256-thread block (8 wave32), full-fp32 WMMA 16x16x4.
//   GEMM1: M=16(B) N=128(Cr) K=512   -> one 16-col tile per wave
//   GEMM2: M=16(B) N=512(C)  K=128   -> four 16-col tiles per wave
// VGPR layouts per CDNA5 ISA 7.12.2 (f32 A 16x4 = 2 VGPRs, B 4x16 = 2 VGPRs,
// D 16x16 = 8 VGPRs).
// ---------------------------------------------------------------------------
__global__ __launch_bounds__(256) void mlp_wmma_kernel(
    const float* __restrict__ att,   // [B, C]
    const float* __restrict__ W1,    // [Cr, C] row-major
    const float* __restrict__ gamma, const float* __restrict__ beta,
    const float* __restrict__ rmean, const float* __restrict__ rvar,
    const float* __restrict__ W2,    // [C, Cr] row-major
    float* __restrict__ a_out) {     // [B, C]
  __shared__ float hT[BB * CR];      // h[b][cr]

  const int lane  = threadIdx.x & 31;
  const int wave  = threadIdx.x >> 5;        // 0..7
  const int m     = lane & 15;               // M row (batch) / N col within tile
  const int khalf = (lane >> 4) * 2;         // lanes 16-31 hold K+2, K+3
  const int rbase = (lane >> 4) * 8;         // D rows: VGPR v -> M = v + rbase

  // -------- GEMM1 + BN + ReLU --------
  {
    const int ncol = wave * 16 + m;          // output channel o in [0,128)
    v8f acc = {};
    for (int k = 0; k < CC; k += 4) {
      v2f a, b;
      a.x = att[m * CC + k + khalf];         // A[M=m, K]
      a.y = att[m * CC + k + khalf + 1];
      b.x = W1[ncol * CC + k + khalf];       // B[K, N=o] = W1[o, K]
      b.y = W1[ncol * CC + k + khalf + 1];
      acc = __builtin_amdgcn_wmma_f32_16x16x4_f32(
          false, a, false, b, (short)0, acc, false, false);
    }
    const float sc = gamma[ncol] * rsqrtf(rvar[ncol] + BN_EPS);
    const float sh = beta[ncol] - rmean[ncol] * sc;
#pragma unroll
    for (int v = 0; v < 8; ++v) {
      float hv = acc[v] * sc + sh;           // D[M = v+rbase, N = ncol]
      hv = hv > 0.0f ? hv : 0.0f;
      hT[(v + rbase) * CR + ncol] = hv;
    }
  }
  __syncthreads();

  // -------- GEMM2 + sigmoid --------
  for (int t = 0; t < 4; ++t) {
    const int ncol = (wave * 4 + t) * 16 + m;  // output channel c in [0,512)
    v8f acc = {};
    for (int k = 0; k < CR; k += 4) {
      v2f a, b;
      a.x = hT[m * CR + k + khalf];          // A[M=m, K]
      a.y = hT[m * CR + k + khalf + 1];
      b.x = W2[ncol * CR + k + khalf];       // B[K, N=c] = W2[c, K]
      b.y = W2[ncol * CR + k + khalf + 1];
      acc = __builtin_amdgcn_wmma_f32_16x16x4_f32(
          false, a, false, b, (short)0, acc, false, false);
    }
#pragma unroll
    for (int v = 0; v < 8; ++v) {
      const float sg = 1.0f / (1.0f + __expf(-acc[v]));
      a_out[(v + rbase) * CC + ncol] = sg;
    }
  }
}

// ---------------------------------------------------------------------------
// Kernel 4: out[b,c,n] = x[b,c,n] * a[b,c].  float4 streaming: x from L2,
// out 134 MB to HBM.
// ---------------------------------------------------------------------------
__global__ __launch_bounds__(256) void scale_kernel(
    const float* __restrict__ x, const float* __restrict__ a,
    float* __restrict__ out) {
  const size_t i4 = (size_t)blockIdx.x * 256 + threadIdx.x;  // float4 index
  const float4* x4 = (const float4*)x;
  float4* o4 = (float4*)out;
  const float av = a[i4 >> 10];  // (i4*4)/N = i4/1024 -> linear b*C + c
  float4 v = x4[i4];
  v.x *= av; v.y *= av; v.z *= av; v.w *= av;
  o4[i4] = v;
}

// ---------------------------------------------------------------------------
extern "C" void kernel_launch(void* const* d_in, const int* in_sizes, int n_in,
                              void* d_out, int out_size, void* d_ws,
                              size_t ws_size, hipStream_t stream) {
  const float* x     = (const float*)d_in[0];  // [16, 512, 4096]
  const float* W1    = (const float*)d_in[1];  // [128, 512]
  const float* gamma = (const float*)d_in[2];  // [128]
  const float* beta  = (const float*)d_in[3];  // [128]
  const float* rmean = (const float*)d_in[4];  // [128]
  const float* rvar  = (const float*)d_in[5];  // [128]
  const float* W2    = (const float*)d_in[6];  // [512, 128]
  float* out = (float*)d_out;                  // [16, 512, 4096]

  // Workspace layout
  float* s_ws   = (float*)d_ws;                            // [16, 4096] 256 KB
  float* att_ws = s_ws + (size_t)BB * NN;                  // [16, 512]   32 KB
  float* a_ws   = att_ws + (size_t)BB * CC;                // [16, 512]   32 KB
  (void)in_sizes; (void)n_in; (void)out_size; (void)ws_size;

  dim3 g1(NN / 256, BB);
  colsum_kernel<<<g1, 256, 0, stream>>>(x, s_ws);

  dim3 g2(CC, BB);
  att_kernel<<<g2, 256, 0, stream>>>(x, s_ws, att_ws);

  mlp_wmma_kernel<<<1, 256, 0, stream>>>(att_ws, W1, gamma, beta, rmean, rvar,
                                         W2, a_ws);

  const int n4 = (BB * CC * NN) / 4;
  scale_kernel<<<n4 / 256, 256, 0, stream>>>(x, a_ws, out);
}